// VF_1752346657369
// MI455X (gfx1250) — compile-verified
//
#include <hip/hip_runtime.h>

typedef __attribute__((ext_vector_type(2))) float v2f;
typedef __attribute__((ext_vector_type(8))) float v8f;

#define NODES 131072
#define EDGES 524288
#define INCH  128
#define HID   256
#define NGR   8192   // NODES / 16 (sum-pool groups)

// ---- deg init + accumulator zero -------------------------------------------
__global__ __launch_bounds__(256) void k_init(float* __restrict__ deg,
                                              float* __restrict__ acc) {
  long long i = (long long)blockIdx.x * blockDim.x + threadIdx.x;
  if (i < (long long)NODES * INCH) acc[i] = 0.0f;
  if (i < NODES) deg[i] = 1.0f;  // self-loop contributes 1
}

__global__ __launch_bounds__(256) void k_deg(const long long* __restrict__ ei,
                                             float* __restrict__ deg) {
  int e = blockIdx.x * blockDim.x + threadIdx.x;
  if (e < EDGES) atomicAdd(&deg[ei[EDGES + e]], 1.0f);  // col = target
}

__global__ __launch_bounds__(256) void k_rsqrt(float* __restrict__ deg) {
  int i = blockIdx.x * blockDim.x + threadIdx.x;
  if (i < NODES) deg[i] = rsqrtf(deg[i]);  // deg >= 1 always
}

// ---- fp32 WMMA GEMM: C = [relu](A[MxK] @ B[KxN] + bias) --------------------
// grid.x = M/16 row-tiles, grid.y = N/128 col-slabs, 256 thr = 8 waves,
// wave w owns 16-col tile w of the slab. Requires K%4==0, N%128==0, M%16==0.
// Inner loop uses pointer-increment addressing so each step is:
//   1x global_load_b64 (A pair) + 2x global_load_b32 (B rows) + v_wmma + adds.
__global__ __launch_bounds__(256) void k_gemm_wmma(
    const float* __restrict__ A, const float* __restrict__ B,
    const float* __restrict__ bias, float* __restrict__ C,
    int M, int N, int K, int relu) {
  const int lane = threadIdx.x & 31;
  const int wave = threadIdx.x >> 5;
  const int row0 = blockIdx.x * 16;
  const int col0 = blockIdx.y * 128 + wave * 16;
  const int mn   = lane & 15;          // row (A frag) / col (B,C frags)
  const int koff = (lane >> 4) * 2;    // K sub-pair per half-wave

  const float* aptr  = A + (size_t)(row0 + mn) * K + koff;   // 8B aligned
  const float* bptr0 = B + (size_t)koff * N + col0 + mn;
  const float* bptr1 = bptr0 + N;
  const size_t bstep = (size_t)4 * N;

  v8f acc = {};
  for (int k = 0; k < K; k += 4) {
    v2f a = *(const v2f*)aptr;
    v2f b;
    b.x = *bptr0;
    b.y = *bptr1;
    acc = __builtin_amdgcn_wmma_f32_16x16x4_f32(
        /*neg_a=*/false, a, /*neg_b=*/false, b,
        /*c_mod=*/(short)0, acc, /*reuse_a=*/false, /*reuse_b=*/false);
    aptr  += 4;
    bptr0 += bstep;
    bptr1 += bstep;
  }

  const float bv = bias ? bias[col0 + mn] : 0.0f;
  const int mb = (lane >> 4) * 8;      // C layout: VGPR v -> M = v + 8*(lane>=16)
  float* cptr = C + (size_t)(row0 + mb) * N + col0 + mn;
  for (int v = 0; v < 8; ++v) {
    float val = acc[v] + bv;
    if (relu) val = fmaxf(val, 0.0f);
    *cptr = val;
    cptr += N;
  }
}

// ---- edge scatter: acc[col] += xw[row] * dinv[row]*dinv[col] ---------------
// 32 lanes per edge, float4 gather (full 512B/edge coalesced), f32 atomics.
__global__ __launch_bounds__(256) void k_scatter(
    const long long* __restrict__ ei, const float* __restrict__ xw,
    const float* __restrict__ dinv, float* __restrict__ acc) {
  long long t = (long long)blockIdx.x * blockDim.x + threadIdx.x;
  int e = (int)(t >> 5);
  int lane = (int)(t & 31);
  if (e >= EDGES) return;
  long long r = ei[e];
  long long c = ei[EDGES + e];
  float w = dinv[r] * dinv[c];
  const float4* src = (const float4*)(xw + r * INCH);
  float4 v = src[lane];
  float* dst = acc + c * INCH + lane * 4;
  atomicAdd(dst + 0, v.x * w);
  atomicAdd(dst + 1, v.y * w);
  atomicAdd(dst + 2, v.z * w);
  atomicAdd(dst + 3, v.w * w);
}

// ---- finalize node feature (self-loop + bias + relu + residual) + sum-pool -
__global__ __launch_bounds__(128) void k_pool(
    const float* __restrict__ acc, const float* __restrict__ xw,
    const float* __restrict__ dinv, const float* __restrict__ state,
    const float* __restrict__ b_gcn, float* __restrict__ pooled) {
  int g = blockIdx.x;
  int c = threadIdx.x;
  float bias = b_gcn[c];
  float s = 0.0f;
  long long base = (long long)g * 16;
  for (int j = 0; j < 16; ++j) {
    long long n = base + j;
    float di = dinv[n];
    float v = acc[n * INCH + c] + xw[n * INCH + c] * (di * di) + bias;
    s += fmaxf(v, 0.0f) + state[n * INCH + c];
  }
  pooled[(long long)g * INCH + c] = s;
}

// ---- head: out[row] = h2[row,:] . W3 + b3 ----------------------------------
__global__ __launch_bounds__(256) void k_head(
    const float* __restrict__ h2, const float* __restrict__ W3,
    const float* __restrict__ b3, float* __restrict__ out) {
  int t = blockIdx.x * blockDim.x + threadIdx.x;
  int row = t >> 5;
  int lane = t & 31;
  if (row >= NGR) return;
  float s = 0.0f;
  for (int i = lane; i < HID; i += 32) s += h2[(long long)row * HID + i] * W3[i];
  for (int off = 16; off > 0; off >>= 1) s += __shfl_down(s, off, 32);
  if (lane == 0) out[row] = s + b3[0];
}

extern "C" void kernel_launch(void* const* d_in, const int* in_sizes, int n_in,
                              void* d_out, int out_size, void* d_ws, size_t ws_size,
                              hipStream_t stream) {
  const float*     state = (const float*)d_in[0];
  const long long* ei    = (const long long*)d_in[1];   // int64 [2, EDGES]
  const float*     W_gcn = (const float*)d_in[2];
  const float*     b_gcn = (const float*)d_in[3];
  const float*     W1    = (const float*)d_in[4];
  const float*     b1    = (const float*)d_in[5];
  const float*     W2    = (const float*)d_in[6];
  const float*     b2    = (const float*)d_in[7];
  const float*     W3    = (const float*)d_in[8];
  const float*     b3    = (const float*)d_in[9];
  float* out = (float*)d_out;

  float* ws     = (float*)d_ws;
  float* deg    = ws;                                   // NODES (becomes dinv)
  float* xw     = deg + NODES;                          // NODES*INCH
  float* acc    = xw + (long long)NODES * INCH;         // NODES*INCH
  float* pooled = acc + (long long)NODES * INCH;        // NGR*INCH
  float* h1     = pooled + (long long)NGR * INCH;       // NGR*HID
  float* h2     = h1 + (long long)NGR * HID;            // NGR*HID

  (void)in_sizes; (void)n_in; (void)out_size; (void)ws_size;

  k_init  <<<(NODES * INCH) / 256, 256, 0, stream>>>(deg, acc);
  k_deg   <<<EDGES / 256,          256, 0, stream>>>(ei, deg);
  k_rsqrt <<<NODES / 256,          256, 0, stream>>>(deg);

  // xw = state @ W_gcn   (bias/relu deferred to pool kernel)
  k_gemm_wmma<<<dim3(NODES / 16, 1), 256, 0, stream>>>(
      state, W_gcn, nullptr, xw, NODES, INCH, INCH, 0);

  k_scatter<<<(EDGES * 32) / 256, 256, 0, stream>>>(ei, xw, deg, acc);

  k_pool<<<NGR, 128, 0, stream>>>(acc, xw, deg, state, b_gcn, pooled);

  // h1 = relu(pooled @ W1 + b1)
  k_gemm_wmma<<<dim3(NGR / 16, HID / 128), 256, 0, stream>>>(
      pooled, W1, b1, h1, NGR, HID, INCH, 1);
  // h2 = relu(h1 @ W2 + b2)
  k_gemm_wmma<<<dim3(NGR / 16, HID / 128), 256, 0, stream>>>(
      h1, W2, b2, h2, NGR, HID, HID, 1);

  k_head<<<(NGR * 32) / 256, 256, 0, stream>>>(h2, W3, b3, out);
}